// AdversarialBlockShift_85993835200844
// MI455X (gfx1250) — compile-verified
//
#include <hip/hip_runtime.h>
#include <hip/hip_bf16.h>

// ---------------------------------------------------------------------------
// AdversarialBlockShift for MI455X (gfx1250, wave32).
//
// out_embeds: embedding gather + 1-D conv (Toeplitz band-GEMM) on the fe
//   slice.  pk is data-derived; a setup kernel compacts the nonzero 4-tap
//   K-chunks so the V_WMMA_F32_16X16X4_F32 GEMM only runs where pk != 0
//   (~5 chunks for the reference delta kernel -> bandwidth-bound).
//   Each wave owns a 16(t) x 64(d) tile: 4 accumulators sharing one A-tile,
//   all bounds checks branchless (clamped load + v_cndmask, no EXEC forks).
// out_ids: cheap index shuffle, written as float (tuple out buffer is f32).
// ---------------------------------------------------------------------------

typedef float v2f __attribute__((ext_vector_type(2)));
typedef float v8f __attribute__((ext_vector_type(8)));

#define B_ 2
#define S_ 4096
#define D_ 512

// Workspace layout (needs ~41KB)
//   hdr[16] ints : 0=nActive 1=ms 2=a0[0] 3=a0[1] 4=fe_start 5=fe_len
//                  6=adv_len 7=ML 8=MR 9=Kp 10=p
#define WS_PK_OFF   64
#define WS_ACT_OFF  (64 + 8192 * 4)
#define ACT_CAP     2048

__device__ __forceinline__ int clampi(int v, int lo, int hi) {
    v = v < lo ? lo : v;
    return v > hi ? hi : v;
}

// ---------------------------------------------------------------------------
// Kernel A: build pk (pad+flip of param), ms, per-batch a0, active chunks.
// ---------------------------------------------------------------------------
__global__ void __launch_bounds__(1024)
setup_kernel(const unsigned char* __restrict__ mask,   // suffix_mask (bool,1B)
             const float* __restrict__ param,
             const int* s_fe_start, const int* s_fe_len, const int* s_adv_len,
             const int* s_ml, const int* s_mr,
             int* __restrict__ hdr, float* __restrict__ pk,
             int* __restrict__ act)
{
    __shared__ int sFirst, sA0[B_];
    __shared__ int sKp, sP, sLP, sPL;
    const int tid = threadIdx.x;
    if (tid == 0) {
        int ml = s_ml[0], mr = s_mr[0];
        int mx = ml > mr ? ml : mr;
        sKp = 2 * mx + 1;
        sP  = mx;
        sLP = (mr > ml) ? (mr - ml) : 0;
        sPL = ml + mr + 1;
        sFirst = 0x7fffffff;
        for (int b = 0; b < B_; ++b) sA0[b] = 0x7fffffff;
        hdr[4] = s_fe_start[0]; hdr[5] = s_fe_len[0]; hdr[6] = s_adv_len[0];
        hdr[7] = ml; hdr[8] = mr; hdr[9] = sKp; hdr[10] = sP;
    }
    __syncthreads();
    const int Kp = sKp, P = sP, LP = sLP, PL = sPL;

    for (int i = tid; i < Kp; i += blockDim.x) {
        int j = Kp - 1 - i - LP;
        float v = (j >= 0 && j < PL) ? param[j] : 0.0f;
        pk[i] = v;
        if (v == 1.0f) atomicMin(&sFirst, i);
    }
    for (int idx = tid; idx < B_ * S_; idx += blockDim.x) {
        if (mask[idx]) {
            int b = idx / S_;
            atomicMin(&sA0[b], idx - b * S_);
        }
    }
    __syncthreads();
    if (tid == 0) {
        int first = (sFirst == 0x7fffffff) ? 0 : sFirst;
        hdr[1] = P - first;                                   // ms
        hdr[2] = (sA0[0] == 0x7fffffff) ? 0 : sA0[0];
        hdr[3] = (sA0[1] == 0x7fffffff) ? 0 : sA0[1];
        // chunk o active iff any pk[m]!=0 for m in [o-15, o+3]
        int n = 0;
        for (int o = 0; o <= Kp + 14; o += 4) {
            int lo = o - 15; if (lo < 0) lo = 0;
            int hi = o + 3;  if (hi > Kp - 1) hi = Kp - 1;
            bool any = false;
            for (int m = lo; m <= hi; ++m)
                if (pk[m] != 0.0f) { any = true; break; }
            if (any && n < ACT_CAP) act[n++] = o;
        }
        hdr[0] = n;
    }
}

// ---------------------------------------------------------------------------
// Kernel B: out_ids (block-shift of the adversarial window) as float.
// ---------------------------------------------------------------------------
__global__ void __launch_bounds__(256)
ids_kernel(const int* __restrict__ ids, const int* __restrict__ hdr,
           float* __restrict__ out_ids)
{
    int g = blockIdx.x * blockDim.x + threadIdx.x;
    if (g >= B_ * S_) return;
    int b = g / S_, j = g % S_;
    int L = hdr[6], ms = hdr[1], a0 = hdr[2 + b];
    int ns = a0 + ms;
    int v;
    if (j >= ns && j < ns + L) {
        int ia = clampi(j - ns, 0, L - 1);
        v = ids[b * S_ + a0 + ia];
    } else {
        int in = clampi((j < ns) ? j : j - L, 0, S_ - L - 1);
        int src = in + ((in >= a0) ? L : 0);
        v = ids[b * S_ + src];
    }
    out_ids[g] = (float)v;
}

// ---------------------------------------------------------------------------
// Kernel C: plain embedding copy for positions outside the fe slice.
// ---------------------------------------------------------------------------
__global__ void __launch_bounds__(256)
copy_kernel(const int* __restrict__ ids, const float* __restrict__ emb,
            const int* __restrict__ hdr, float* __restrict__ out)
{
    const int d4 = D_ / 4;
    long g = (long)blockIdx.x * blockDim.x + threadIdx.x;
    long t_lin = g / d4;
    int  dq    = (int)(g % d4);
    if (t_lin >= (long)B_ * S_) return;
    int b = (int)(t_lin / S_), t = (int)(t_lin % S_);
    int fs = hdr[4], fl = hdr[5];
    if (t >= fs && t < fs + fl) return;
    int tok = ids[b * S_ + t];
    float4 v = ((const float4*)(emb + (long)tok * D_))[dq];
    ((float4*)(out + ((long)(b * S_) + t) * D_))[dq] = v;
}

// ---------------------------------------------------------------------------
// Kernel D: Toeplitz band-GEMM via V_WMMA_F32_16X16X4_F32.
// One wave32 per 16(t) x 64(d) tile; 4 accumulators share one A-tile.
//   A (16x4):  A[i][j] = pk[o + j - i]
//   B (4x16):  B[k][n] = emb[ids[fs + t0+o-p+k]][d0+n]  (zero-padded edges)
// Layouts per ISA 7.12.2 (wave32): A lanes0-15 K=0/1, lanes16-31 K=2/3;
// B mirrors the C/D half-split; C/D v8f: VGPR r -> (M=r|r+8, N=lane&15).
// All OOB handling is branchless: clamped address + select-0 (no EXEC forks),
// so EXEC is all-ones at every WMMA issue.
// ---------------------------------------------------------------------------
__global__ void __launch_bounds__(256)
conv_kernel(const int* __restrict__ ids, const float* __restrict__ emb,
            const int* __restrict__ hdr, const float* __restrict__ pk,
            const int* __restrict__ act, float* __restrict__ out)
{
    const int DG = D_ / 64;           // 8 d-groups of 64 columns
    const int TT = S_ / 16;           // 256 t-tiles (covers fe_len)
    int wave = (blockIdx.x * blockDim.x + threadIdx.x) >> 5;
    int lane = threadIdx.x & 31;

    int dg   = wave % DG;
    int rest = wave / DG;
    int tt   = rest % TT;
    int b    = rest / TT;
    if (b >= B_) return;                          // wave-uniform

    const int fs = hdr[4], T = hdr[5], Kp = hdr[9], P = hdr[10];
    const int nAct = hdr[0];
    const int t0 = tt * 16;
    if (t0 >= T) return;                          // wave-uniform
    const int d0 = dg * 64;

    const int row   = lane & 15;                  // A row / B-C column
    const int khalf = (lane >> 4) * 2;            // 0 | 2
    const int* idrow = ids + b * S_ + fs;

    v8f c0 = {0.f,0.f,0.f,0.f,0.f,0.f,0.f,0.f};
    v8f c1 = c0, c2 = c0, c3 = c0;

    for (int a = 0; a < nAct; ++a) {
        int o = act[a];                           // wave-uniform
        // --- A tile (branchless: clamped pk load + select) ---
        int m0 = o + khalf - row;
        int m1 = m0 + 1;
        float pa0 = pk[clampi(m0, 0, Kp - 1)];
        float pa1 = pk[clampi(m1, 0, Kp - 1)];
        v2f av;
        av.x = (m0 >= 0 && m0 < Kp) ? pa0 : 0.0f;
        av.y = (m1 >= 0 && m1 < Kp) ? pa1 : 0.0f;
        // --- B tiles: gather 2 rows x 64 cols (branchless) ---
        int rbase = t0 + o - P;
        int r0 = rbase + khalf, r1 = r0 + 1;
        bool v0 = (r0 >= 0) && (r0 < T);
        bool v1 = (r1 >= 0) && (r1 < T);
        const float* p0 = emb + (long)idrow[clampi(r0, 0, T - 1)] * D_ + d0 + row;
        const float* p1 = emb + (long)idrow[clampi(r1, 0, T - 1)] * D_ + d0 + row;
        float b00 = p0[0], b01 = p0[16], b02 = p0[32], b03 = p0[48];
        float b10 = p1[0], b11 = p1[16], b12 = p1[32], b13 = p1[48];
        v2f bv0, bv1, bv2, bv3;
        bv0.x = v0 ? b00 : 0.0f;  bv0.y = v1 ? b10 : 0.0f;
        bv1.x = v0 ? b01 : 0.0f;  bv1.y = v1 ? b11 : 0.0f;
        bv2.x = v0 ? b02 : 0.0f;  bv2.y = v1 ? b12 : 0.0f;
        bv3.x = v0 ? b03 : 0.0f;  bv3.y = v1 ? b13 : 0.0f;
        // 4 independent WMMAs sharing A (pipeline into the XDL)
        c0 = __builtin_amdgcn_wmma_f32_16x16x4_f32(false, av, false, bv0,
                                                   (short)0, c0, false, false);
        c1 = __builtin_amdgcn_wmma_f32_16x16x4_f32(false, av, false, bv1,
                                                   (short)0, c1, false, false);
        c2 = __builtin_amdgcn_wmma_f32_16x16x4_f32(false, av, false, bv2,
                                                   (short)0, c2, false, false);
        c3 = __builtin_amdgcn_wmma_f32_16x16x4_f32(false, av, false, bv3,
                                                   (short)0, c3, false, false);
    }

    // Store 16x64 tile (mask rows past T)
    int n     = lane & 15;
    int mbase = (lane >> 4) * 8;
#pragma unroll
    for (int r = 0; r < 8; ++r) {
        int t = t0 + mbase + r;
        if (t < T) {
            float* dst = out + ((long)(b * S_) + fs + t) * D_ + d0 + n;
            dst[0]  = c0[r];
            dst[16] = c1[r];
            dst[32] = c2[r];
            dst[48] = c3[r];
        }
    }
}

// ---------------------------------------------------------------------------
extern "C" void kernel_launch(void* const* d_in, const int* in_sizes, int n_in,
                              void* d_out, int out_size, void* d_ws, size_t ws_size,
                              hipStream_t stream)
{
    const int*           ids   = (const int*)d_in[0];
    const unsigned char* mask  = (const unsigned char*)d_in[1];  // jnp bool
    const float*         param = (const float*)d_in[2];
    const float*         emb   = (const float*)d_in[3];
    const int* s_fe_start = (const int*)d_in[4];
    const int* s_fe_len   = (const int*)d_in[5];
    const int* s_adv_len  = (const int*)d_in[6];
    const int* s_ml       = (const int*)d_in[7];
    const int* s_mr       = (const int*)d_in[8];

    int*   hdr = (int*)d_ws;
    float* pk  = (float*)((char*)d_ws + WS_PK_OFF);
    int*   act = (int*)((char*)d_ws + WS_ACT_OFF);

    float* out_emb = (float*)d_out;                     // B*S*D floats
    float* out_ids = out_emb + (long)B_ * S_ * D_;      // B*S ids-as-floats

    setup_kernel<<<1, 1024, 0, stream>>>(mask, param,
                                         s_fe_start, s_fe_len, s_adv_len,
                                         s_ml, s_mr, hdr, pk, act);
    ids_kernel<<<(B_ * S_ + 255) / 256, 256, 0, stream>>>(ids, hdr, out_ids);
    {
        long total = (long)B_ * S_ * (D_ / 4);
        copy_kernel<<<(unsigned)((total + 255) / 256), 256, 0, stream>>>(
            ids, emb, hdr, out_emb);
    }
    {
        long waves  = (long)B_ * (S_ / 16) * (D_ / 64);   // 4096 waves
        long blocks = (waves * 32 + 255) / 256;           // 8 waves / block
        conv_kernel<<<(unsigned)blocks, 256, 0, stream>>>(
            ids, emb, hdr, pk, act, out_emb);
    }
}